// VideoGameDnn_88261577933008
// MI455X (gfx1250) — compile-verified
//
#include <hip/hip_runtime.h>
#include <hip/hip_bf16.h>
#include <math.h>

// ---------------------------------------------------------------------------
// DLRM-style fused kernel for MI455X (gfx1250, wave32).
//   pooled embeddings (segment-sum by id/50000)  ->  832->64 GEMM (WMMA f32)
//   -> ReLU -> 64->16 GEMM (WMMA f32) -> ReLU -> 16->1 dot -> sigmoid
// One block of 256 threads (8 waves) processes 16 batch rows.
// ---------------------------------------------------------------------------

#define N_FEATURES       26
#define VOCAB_PER_FEAT   50000
#define SLOTS_PER_FEAT   8
#define P_IDS            (N_FEATURES * SLOTS_PER_FEAT)   // 208 ids per row
#define EMBED            32
#define D_FEATS          (N_FEATURES * EMBED)            // 832
#define H0_DIM           64
#define H1_DIM           16
#define ROWS_PER_BLOCK   16

// LDS strides chosen so A-fragment reads (lane stride = row stride) hit
// distinct banks: 833 % 64 == 1, 65 % 64 == 1.
#define FSTRIDE 833   // feats row stride in dwords
#define HSTRIDE 65    // h0 row stride in dwords

typedef float v2f __attribute__((ext_vector_type(2)));
typedef float v8f __attribute__((ext_vector_type(8)));

__global__ __launch_bounds__(256)
void VideoGameDnn_fused_kernel(const int*   __restrict__ x,      // [B, 208]
                               const float* __restrict__ table,  // [VOCAB, 32]
                               const float* __restrict__ W0,     // [832, 64]
                               const float* __restrict__ b0,     // [64]
                               const float* __restrict__ W1,     // [64, 16]
                               const float* __restrict__ b1,     // [16]
                               const float* __restrict__ W2,     // [16, 1]
                               float*       __restrict__ out)    // [B]
{
    __shared__ float feats[ROWS_PER_BLOCK * FSTRIDE];  // 16 x 832 pooled feats
    __shared__ float h0s  [ROWS_PER_BLOCK * HSTRIDE];  // 16 x 64 hidden
    __shared__ float h1s  [ROWS_PER_BLOCK * H1_DIM];   // 16 x 16 hidden

    const int tid    = threadIdx.x;
    const int wave   = tid >> 5;          // 0..7
    const int lane   = tid & 31;          // 0..31
    const int rowTop = blockIdx.x * ROWS_PER_BLOCK;

    // ---- zero the pooled-feature LDS tile --------------------------------
    for (int i = tid; i < ROWS_PER_BLOCK * FSTRIDE; i += 256) feats[i] = 0.0f;
    __syncthreads();

    // ---- stage 1: embedding gather + segment-sum pooling -----------------
    // wave w owns rows 2w and 2w+1; lane = embedding column (coalesced 128B).
    // Each lane touches only column `lane` of the LDS tile, and LDS ops from
    // one wave are in-order, so the read-modify-write below is race-free.
    for (int r = 0; r < 2; ++r) {
        const int row = wave * 2 + r;                 // 0..15
        const long gbase = (long)(rowTop + row) * P_IDS;
        for (int p = 0; p < P_IDS; ++p) {
            const int id  = x[gbase + p];             // uniform across wave
            if (p + 1 < P_IDS) {
                const int idn = x[gbase + p + 1];
                __builtin_prefetch(table + (size_t)idn * EMBED, 0, 1);
            }
            const int cat = id / VOCAB_PER_FEAT;      // 0..25
            const float v = table[(size_t)id * EMBED + lane];
            feats[row * FSTRIDE + cat * EMBED + lane] += v;
        }
    }
    __syncthreads();

    // ---- stage 2: feats[16x832] @ W0[832x64] + b0, ReLU ------------------
    // 4 waves, each owns one N=16 slice; 208 k-steps of v_wmma_f32_16x16x4_f32.
    if (wave < 4) {
        const int n0 = wave * 16;
        const int m  = lane & 15;                  // A/B row within fragment
        const int kk = (lane >> 4) << 1;           // K offset 0 or 2 (lane[4])
        const int n  = n0 + (lane & 15);
        v8f c = {};
#pragma unroll 4
        for (int k0 = 0; k0 < D_FEATS / 4; ++k0) {
            const int kb = k0 * 4 + kk;
            v2f a, b;
            a.x = feats[m * FSTRIDE + kb];
            a.y = feats[m * FSTRIDE + kb + 1];
            b.x = W0[(size_t)kb * H0_DIM + n];
            b.y = W0[(size_t)(kb + 1) * H0_DIM + n];
            c = __builtin_amdgcn_wmma_f32_16x16x4_f32(
                    false, a, false, b, (short)0, c, false, false);
        }
        // C layout: VGPR r -> rows r (lanes 0-15) and r+8 (lanes 16-31)
        const int mrow = (lane < 16) ? 0 : 8;
#pragma unroll
        for (int r = 0; r < 8; ++r) {
            float v = c[r] + b0[n];
            h0s[(r + mrow) * HSTRIDE + n] = v > 0.0f ? v : 0.0f;
        }
    }
    __syncthreads();

    // ---- stage 3: h0[16x64] @ W1[64x16] + b1, ReLU; then @ W2; sigmoid ---
    if (wave == 0) {
        const int m  = lane & 15;
        const int kk = (lane >> 4) << 1;
        const int n  = lane & 15;
        v8f c = {};
#pragma unroll
        for (int k0 = 0; k0 < H0_DIM / 4; ++k0) {
            const int kb = k0 * 4 + kk;
            v2f a, b;
            a.x = h0s[m * HSTRIDE + kb];
            a.y = h0s[m * HSTRIDE + kb + 1];
            b.x = W1[kb * H1_DIM + n];
            b.y = W1[(kb + 1) * H1_DIM + n];
            c = __builtin_amdgcn_wmma_f32_16x16x4_f32(
                    false, a, false, b, (short)0, c, false, false);
        }
        const int mrow = (lane < 16) ? 0 : 8;
#pragma unroll
        for (int r = 0; r < 8; ++r) {
            float v = c[r] + b1[n];
            h1s[(r + mrow) * H1_DIM + n] = v > 0.0f ? v : 0.0f;
        }
        // 16 -> 1 projection + sigmoid on lanes 0..15 (in-order LDS per wave)
        if (lane < 16) {
            float acc = 0.0f;
#pragma unroll
            for (int j = 0; j < H1_DIM; ++j)
                acc += h1s[lane * H1_DIM + j] * W2[j];
            out[rowTop + lane] = 1.0f / (1.0f + expf(-acc));
        }
    }
}

extern "C" void kernel_launch(void* const* d_in, const int* in_sizes, int n_in,
                              void* d_out, int out_size, void* d_ws, size_t ws_size,
                              hipStream_t stream) {
    (void)in_sizes; (void)n_in; (void)d_ws; (void)ws_size;
    const int*   x     = (const int*)  d_in[0];
    const float* table = (const float*)d_in[1];
    const float* W0    = (const float*)d_in[2];
    const float* b0    = (const float*)d_in[3];
    const float* W1    = (const float*)d_in[4];
    const float* b1    = (const float*)d_in[5];
    const float* W2    = (const float*)d_in[6];
    float*       out   = (float*)d_out;

    const int batch  = out_size;                       // 16384
    const int blocks = (batch + ROWS_PER_BLOCK - 1) / ROWS_PER_BLOCK;
    VideoGameDnn_fused_kernel<<<blocks, 256, 0, stream>>>(
        x, table, W0, b0, W1, b1, W2, out);
}